// TFMambaBlock_42966852829385
// MI455X (gfx1250) — compile-verified
//
#include <hip/hip_runtime.h>
#include <hip/hip_bf16.h>

// ---------------------------------------------------------------------------
// TF-Mamba (bidirectional time + frequency Mamba) for MI455X / gfx1250.
// fp32 end-to-end; all dense projections run on V_WMMA_F32_16X16X4_F32.
// ---------------------------------------------------------------------------

typedef __attribute__((ext_vector_type(2))) float v2f;
typedef __attribute__((ext_vector_type(8))) float v8f;

#define CC   64
#define LL   128     // sequence length (t or f)
#define SS   128     // number of sequences (f or t)
#define DD   256     // expanded dim
#define NN   16      // state dim
#define RR   (SS*LL) // 16384 rows

__device__ __forceinline__ float silu_f(float v) {
    return v * (1.0f / (1.0f + __expf(-v)));
}

// ---------------- layout kernels ----------------

// x (c,t,f) -> xt[(f*L + t)*64 + c]
__global__ void transpose_in_kernel(const float* __restrict__ x, float* __restrict__ xt) {
    int idx = blockIdx.x * blockDim.x + threadIdx.x;   // over 64*128*128
    if (idx >= CC * LL * SS) return;
    int f = idx & 127;
    int t = (idx >> 7) & 127;
    int c = idx >> 14;
    xt[((size_t)(f * LL + t)) * CC + c] = x[idx];
}

// xo[(f*T + t)] -> xt[(t*F + f)]   (time-major -> freq-major sequences)
__global__ void relayout_tf_kernel(const float* __restrict__ src, float* __restrict__ dst) {
    int idx = blockIdx.x * blockDim.x + threadIdx.x;   // over R*64
    if (idx >= RR * CC) return;
    int c = idx % CC;
    int row = idx / CC;
    int t = row % LL;       // position inside old sequence
    int f = row / LL;       // old sequence id
    dst[((size_t)(t * SS + f)) * CC + c] = src[idx];
}

// xo[(t*F + f)*64 + c] -> out[(c*T + t)*F + f]
__global__ void transpose_out_kernel(const float* __restrict__ src, float* __restrict__ out) {
    int idx = blockIdx.x * blockDim.x + threadIdx.x;   // over R*64
    if (idx >= RR * CC) return;
    int c = idx % CC;
    int row = idx / CC;
    int f = row % SS;
    int t = row / SS;
    out[((size_t)(c * LL + t)) * SS + f] = src[idx];
}

// reverse each sequence in time: dst[s][t] = src[s][L-1-t]
__global__ void reverse_seq_kernel(const float* __restrict__ src, float* __restrict__ dst) {
    int idx = blockIdx.x * blockDim.x + threadIdx.x;   // over R*64
    if (idx >= RR * CC) return;
    int c = idx % CC;
    int row = idx / CC;
    int t = row % LL;
    int s = row / LL;
    dst[idx] = src[((size_t)(s * LL + (LL - 1 - t))) * CC + c];
}

// ---------------- rms-norm: one wave32 per 64-channel row ----------------
__global__ void rmsnorm_kernel(const float* __restrict__ x, const float* __restrict__ w,
                               float* __restrict__ y, int rows) {
    int lane = threadIdx.x & 31;
    int wave = threadIdx.x >> 5;
    int row = blockIdx.x * (blockDim.x >> 5) + wave;
    if (row >= rows) return;
    const float* xr = x + (size_t)row * CC;
    float a = xr[lane];
    float b = xr[lane + 32];
    float ss = a * a + b * b;
    #pragma unroll
    for (int off = 16; off >= 1; off >>= 1) ss += __shfl_xor(ss, off, 32);
    float scale = rsqrtf(ss * (1.0f / 64.0f) + 1e-5f);
    y[(size_t)row * CC + lane]      = a * scale * w[lane];
    y[(size_t)row * CC + lane + 32] = b * scale * w[lane + 32];
}

// ---------------- fp32 WMMA GEMM ----------------
// C[orow, coff+n] (+bias +resid) = sum_k A[row,k] * B(k,n)
//   WT=1    : B(k,n) = W[n*K + k]   (x @ W^T, W is (N,K))
//   WT=0    : B(k,n) = W[k*N + n]   (x @ W,   W is (K,N))
//   NT      : 16-wide N tiles per wave (one A fragment feeds NT WMMAs per k-step)
//   MASKED  : 1 only when N is not a multiple of the wave tiling (x_proj, N=36);
//             out-of-range columns are clamped to valid addresses + zero-masked.
//   rev=1   : output row s*L + (L-1-t) for input row s*L+t (backward direction)
//   resid   : (nullable) 64-wide residual added at output row.
// All A/B accesses are integer-offset GEPs on the kernel-arg pointers so the
// compiler keeps them in the GLOBAL address space (global_load_b64, LOADcnt
// only) instead of falling back to FLAT (which also ticks DScnt).
// EXEC stays all-1s through every WMMA (no divergent control flow in the loop).
template<int WT, int NT, int MASKED>
__global__ void gemm_wmma_f32(const float* __restrict__ A, const float* __restrict__ W,
                              float* __restrict__ C,
                              int N, int K, int ldc, int coff,
                              const float* __restrict__ bias,
                              const float* __restrict__ resid, int rev) {
    const int lane = threadIdx.x & 31;
    const int wave = threadIdx.x >> 5;
    const int m0 = blockIdx.x * 16;
    const int n0 = (blockIdx.y * 4 + wave) * (16 * NT);

    const int trow = lane & 15;                  // A row / B,C column within tile
    const int kh   = (lane >> 4) << 1;           // 0 or 2 (K sub-offset per ISA layout)

    v8f acc[NT];
    #pragma unroll
    for (int j = 0; j < NT; ++j) acc[j] = (v8f){0.f,0.f,0.f,0.f,0.f,0.f,0.f,0.f};

    // integer base offsets (not pointers) keep loads in the global addrspace
    size_t aoff = (size_t)(m0 + trow) * K + kh;

    size_t boff[NT];
    float bm[NT];
    #pragma unroll
    for (int j = 0; j < NT; ++j) {
        int nidx = n0 + j * 16 + trow;
        bm[j] = 1.f;
        int nc = nidx;
        if (MASKED) {
            bm[j] = (nidx < N) ? 1.f : 0.f;
            nc = (nidx < N) ? nidx : (N - 1);    // clamped: load stays in-bounds
        }
        boff[j] = WT ? ((size_t)nc * K + kh)
                     : ((size_t)kh * N + nc);
    }
    const size_t bsec  = WT ? 1 : (size_t)N;     // b.x -> b.y element offset
    const size_t bstep = WT ? 4 : 4 * (size_t)N; // offset step per k += 4
    size_t ko = 0;

    for (int k = 0; k < K; k += 4) {
        v2f a;
        a.x = A[aoff];
        a.y = A[aoff + 1];
        aoff += 4;
        v2f b[NT];
        #pragma unroll
        for (int j = 0; j < NT; ++j) {           // load clause first ...
            b[j].x = W[boff[j] + ko];
            b[j].y = W[boff[j] + ko + bsec];
            if (MASKED) { b[j].x *= bm[j]; b[j].y *= bm[j]; }
        }
        ko += bstep;
        #pragma unroll
        for (int j = 0; j < NT; ++j)             // ... then WMMA burst
            acc[j] = __builtin_amdgcn_wmma_f32_16x16x4_f32(false, a, false, b[j],
                                                           (short)0, acc[j], false, false);
    }

    const int mbase = m0 + ((lane >> 4) << 3);   // rows 0..7 or 8..15 of the tile
    #pragma unroll
    for (int j = 0; j < NT; ++j) {
        int nidx = n0 + j * 16 + trow;
        if (MASKED && nidx >= N) continue;
        float bval = bias ? bias[nidx] : 0.f;
        #pragma unroll
        for (int r = 0; r < 8; ++r) {
            int mm = mbase + r;
            int orow = mm;
            if (rev) {
                int s = mm >> 7;                 // LL == 128
                int t = mm & 127;
                orow = (s << 7) + (127 - t);
            }
            float v = acc[j][r] + bval;
            if (resid) v += resid[(size_t)orow * CC + nidx];
            C[(size_t)orow * ldc + coff + nidx] = v;
        }
    }
}

// ---------------- causal depthwise conv (K=4) + SiLU ----------------
// reads xc = xz[:, 0:256], writes xcs[row*256 + dch]
__global__ void dwconv_silu_kernel(const float* __restrict__ xz,
                                   const float* __restrict__ cw,
                                   const float* __restrict__ cb,
                                   float* __restrict__ out) {
    int idx = blockIdx.x * blockDim.x + threadIdx.x;   // over R*256
    if (idx >= RR * DD) return;
    int dch = idx & (DD - 1);
    int row = idx >> 8;
    int t = row % LL;
    int s = row / LL;
    float acc = cb[dch];
    #pragma unroll
    for (int k = 0; k < 4; ++k) {
        int tt = t - 3 + k;
        if (tt >= 0)
            acc += cw[dch * 4 + k] * xz[((size_t)(s * LL + tt)) * 512 + dch];
    }
    out[idx] = silu_f(acc);
}

// ---------------- dt projection (r=4) + softplus ----------------
__global__ void dtproj_kernel(const float* __restrict__ dbl,
                              const float* __restrict__ dtw,
                              const float* __restrict__ dtb,
                              float* __restrict__ dtv) {
    int idx = blockIdx.x * blockDim.x + threadIdx.x;   // over R*256
    if (idx >= RR * DD) return;
    int dch = idx & (DD - 1);
    int row = idx >> 8;
    const float* dr = dbl + (size_t)row * 48;
    float v = dtb[dch];
    #pragma unroll
    for (int q = 0; q < 4; ++q) v += dr[q] * dtw[dch * 4 + q];
    dtv[idx] = (v > 20.f) ? v : log1pf(__expf(v));      // softplus
}

// ---------------- selective scan: one workgroup per sequence ----------------
// state h[d=threadIdx.x][n=0..15] kept in registers; B_t/C_t broadcast via LDS.
// Fuses  y = scan + D*xc  and the SiLU(z) gate.
__global__ void scan_kernel(const float* __restrict__ xcs,
                            const float* __restrict__ dtv,
                            const float* __restrict__ dbl,
                            const float* __restrict__ Alog,
                            const float* __restrict__ Dp,
                            const float* __restrict__ xz,   // z at column 256..511
                            float* __restrict__ ysc) {
    int s = blockIdx.x;        // sequence id
    int dch = threadIdx.x;     // 0..255
    float Aj[NN], h[NN];
    #pragma unroll
    for (int j = 0; j < NN; ++j) {
        Aj[j] = -__expf(Alog[(size_t)dch * NN + j]);
        h[j] = 0.f;
    }
    float Dv = Dp[dch];
    __shared__ float Bs[NN], Cs[NN];

    for (int t = 0; t < LL; ++t) {
        size_t row = (size_t)s * LL + t;
        if (threadIdx.x < 32) {
            float v = dbl[row * 48 + 4 + threadIdx.x];   // cols 4..19 = B, 20..35 = C
            if (threadIdx.x < 16) Bs[threadIdx.x] = v;
            else                  Cs[threadIdx.x - 16] = v;
        }
        __syncthreads();
        float dt = dtv[row * DD + dch];
        float xv = xcs[row * DD + dch];
        float dtx = dt * xv;
        float y = 0.f;
        #pragma unroll
        for (int j = 0; j < NN; ++j) {
            h[j] = h[j] * __expf(dt * Aj[j]) + dtx * Bs[j];
            y += h[j] * Cs[j];
        }
        float zv = xz[row * 512 + DD + dch];
        ysc[row * DD + dch] = (y + Dv * xv) * silu_f(zv);
        __syncthreads();
    }
}

// ---------------------------------------------------------------------------

extern "C" void kernel_launch(void* const* d_in, const int* in_sizes, int n_in,
                              void* d_out, int out_size, void* d_ws, size_t ws_size,
                              hipStream_t stream) {
    const float* x    = (const float*)d_in[0];
    const float* nw   = (const float*)d_in[1];   // (4,64)
    const float* ipw  = (const float*)d_in[2];   // (4,512,64)
    const float* cw   = (const float*)d_in[3];   // (4,256,4)
    const float* cb   = (const float*)d_in[4];   // (4,256)
    const float* xpw  = (const float*)d_in[5];   // (4,36,256)
    const float* dtw  = (const float*)d_in[6];   // (4,256,4)
    const float* dtb  = (const float*)d_in[7];   // (4,256)
    const float* Alog = (const float*)d_in[8];   // (4,256,16)
    const float* Dp   = (const float*)d_in[9];   // (4,256)
    const float* opw  = (const float*)d_in[10];  // (4,64,256)
    const float* tlw  = (const float*)d_in[11];  // (128,64)
    const float* tlb  = (const float*)d_in[12];  // (64)
    const float* flw  = (const float*)d_in[13];  // (128,64)
    const float* flb  = (const float*)d_in[14];  // (64)
    float* out = (float*)d_out;

    float* ws   = (float*)d_ws;
    float* xt   = ws;                        // R*64   stage input (residual source)
    float* xrev = xt   + (size_t)RR * 64;    // R*64
    float* xn   = xrev + (size_t)RR * 64;    // R*64
    float* xz   = xn   + (size_t)RR * 64;    // R*512  [xc | z]
    float* xcs  = xz   + (size_t)RR * 512;   // R*256  conv+silu
    float* dbl  = xcs  + (size_t)RR * 256;   // R*48   [dt_raw(4) | B(16) | C(16) | pad]
    float* dtv  = dbl  + (size_t)RR * 48;    // R*256
    float* ysc  = dtv  + (size_t)RR * 256;   // R*256
    float* ycat = ysc  + (size_t)RR * 256;   // R*128  [fwd(64) | bwd(64)]
    float* xo   = ycat + (size_t)RR * 128;   // R*64   stage output

    const int ELT = (RR * CC + 255) / 256;   // grid for R*64 elementwise kernels
    const int EL2 = (RR * DD + 255) / 256;   // grid for R*256 elementwise kernels

    auto run_bidir = [&](const float* linw, const float* linb, int i_f, int i_b) {
        for (int dir = 0; dir < 2; ++dir) {
            int pi = (dir == 0) ? i_f : i_b;
            const float* src = xt;
            if (dir == 1) {
                reverse_seq_kernel<<<ELT, 256, 0, stream>>>(xt, xrev);
                src = xrev;
            }
            rmsnorm_kernel<<<RR / 8, 256, 0, stream>>>(src, nw + pi * 64, xn, RR);
            // in_proj: [R,64] x (512,64)^T -> xz [R,512]; 4 N-tiles per wave
            gemm_wmma_f32<1, 4, 0><<<dim3(RR / 16, 2), 128, 0, stream>>>(
                xn, ipw + (size_t)pi * 512 * 64, xz,
                512, 64, 512, 0, nullptr, nullptr, 0);
            dwconv_silu_kernel<<<EL2, 256, 0, stream>>>(
                xz, cw + (size_t)pi * 256 * 4, cb + (size_t)pi * 256, xcs);
            // x_proj: [R,256] x (36,256)^T -> dbl [R,36] (ldc 48; cols >=36 masked)
            gemm_wmma_f32<1, 1, 1><<<dim3(RR / 16, 1), 128, 0, stream>>>(
                xcs, xpw + (size_t)pi * 36 * 256, dbl,
                36, 256, 48, 0, nullptr, nullptr, 0);
            dtproj_kernel<<<EL2, 256, 0, stream>>>(
                dbl, dtw + (size_t)pi * 256 * 4, dtb + (size_t)pi * 256, dtv);
            scan_kernel<<<SS, DD, 0, stream>>>(
                xcs, dtv, dbl, Alog + (size_t)pi * 256 * 16, Dp + (size_t)pi * 256,
                xz, ysc);
            // out_proj: [R,256] x (64,256)^T -> ycat[:, dir*64 : dir*64+64]
            //   fwd adds xt residual; bwd stores at reversed t with xt residual.
            gemm_wmma_f32<1, 1, 0><<<dim3(RR / 16, 1), 128, 0, stream>>>(
                ysc, opw + (size_t)pi * 64 * 256, ycat,
                64, 256, 128, dir * 64, nullptr, xt, dir);
        }
        // mixing: [R,128] x (128,64) + bias + xt residual -> xo [R,64]
        gemm_wmma_f32<0, 1, 0><<<dim3(RR / 16, 1), 128, 0, stream>>>(
            ycat, linw, xo, 64, 128, 64, 0, linb, xt, 0);
    };

    // ---- stage A: time-direction mamba (sequences indexed by f) ----
    transpose_in_kernel<<<ELT, 256, 0, stream>>>(x, xt);
    run_bidir(tlw, tlb, 0, 1);

    // ---- stage B: frequency-direction mamba (sequences indexed by t) ----
    relayout_tf_kernel<<<ELT, 256, 0, stream>>>(xo, xt);
    run_bidir(flw, flb, 2, 3);

    // ---- final layout: (t,f,c) -> (c,t,f) ----
    transpose_out_kernel<<<ELT, 256, 0, stream>>>(xo, out);
}